// CrossLoRALinear_31980326486354
// MI455X (gfx1250) — compile-verified
//
#include <hip/hip_runtime.h>

// ---------------------------------------------------------------------------
// LoRA linear: out = x @ W^T + b + (x@A^T)@B^T, fp32.
//   Step 1: W_eff = W + B@A   (rank-16 update folded into weight, 16 MiB d_ws)
//   Step 2: out = x @ W_eff^T + b   (fp32 WMMA GEMM)
// M=16384, N=2048, K=2048. HBM floor ~300 MB @ 23.3 TB/s ~= 13 us; needs only
// ~11 TFLOP/s fp32 -> V_WMMA_F32_16X16X4_F32 keeps it bandwidth-bound while
// matching the fp32 reference exactly.
// 256x256 block tiles (16 waves) halve L2 re-read traffic vs 128x128; global
// operand tiles are fetched with CDNA5 async LDS loads
// (global_load_async_to_lds_b128, ASYNCcnt) -- no VGPR staging.
// ---------------------------------------------------------------------------

typedef __attribute__((ext_vector_type(2))) float v2f;
typedef __attribute__((ext_vector_type(8))) float v8f;

#define M_DIM 16384
#define N_DIM 2048
#define K_DIM 2048
#define R_DIM 16

#define BM 256
#define BN 256
#define BK 16
#define BKP 20       // padded LDS row stride: 20*m mod 64 distinct (m=0..15), 16B aligned
#define THREADS 512  // 16 waves -> 4x4 grid of 64x64 wave tiles

// --------------------------- W_eff = W + B @ A ------------------------------
__global__ __launch_bounds__(256) void weff_kernel(const float* __restrict__ W,
                                                   const float* __restrict__ A,
                                                   const float* __restrict__ Bm,
                                                   float* __restrict__ Weff) {
  int idx4 = blockIdx.x * 256 + threadIdx.x;   // one float4 per thread
  int o  = idx4 / (K_DIM / 4);                 // uniform within a block
  int d4 = idx4 % (K_DIM / 4);
  float4 w = ((const float4*)(W + (size_t)o * K_DIM))[d4];
#pragma unroll
  for (int r = 0; r < R_DIM; ++r) {
    float  br = Bm[o * R_DIM + r];
    float4 a  = ((const float4*)(A + (size_t)r * K_DIM))[d4];
    w.x += br * a.x; w.y += br * a.y; w.z += br * a.z; w.w += br * a.w;
  }
  ((float4*)(Weff + (size_t)o * K_DIM))[d4] = w;
}

// ------------------- CDNA5 async global->LDS helpers ------------------------
__device__ __forceinline__ void async_load_b128(const float* g, const float* lds) {
  // LDS byte address = low 32 bits of the generic pointer (aperture holds hi bits)
  unsigned int       l  = (unsigned int)(unsigned long long)(uintptr_t)lds;
  unsigned long long ga = (unsigned long long)(uintptr_t)g;
  asm volatile("global_load_async_to_lds_b128 %0, %1, off"
               :: "v"(l), "v"(ga) : "memory");
}

__device__ __forceinline__ void wait_async0() {
#if __has_builtin(__builtin_amdgcn_s_wait_asynccnt)
  __builtin_amdgcn_s_wait_asynccnt(0);
#else
  asm volatile("s_wait_asynccnt 0x0" ::: "memory");
#endif
}

// --------------------- out = x @ W_eff^T + b (fp32 WMMA) --------------------
__global__ __launch_bounds__(THREADS) void lora_gemm_kernel(const float* __restrict__ X,
                                                            const float* __restrict__ Weff,
                                                            const float* __restrict__ bias,
                                                            float* __restrict__ out) {
  __shared__ float xs[2][BM][BKP];   // x tile: M rows, K contiguous  (40 KB)
  __shared__ float wt[2][BN][BKP];   // W tile: N rows, K contiguous  (40 KB)

  const int tid   = threadIdx.x;
  const int lane  = tid & 31;
  const int wid   = tid >> 5;            // 16 waves -> 4x4 grid of 64x64 tiles
  const int waveM = (wid >> 2) * 64;
  const int waveN = (wid & 3) * 64;
  const int m0 = blockIdx.y * BM;
  const int n0 = blockIdx.x * BN;
  const int l15   = lane & 15;
  const int hi    = lane >> 4;           // 0: lanes 0-15, 1: lanes 16-31
  const int khalf = hi << 1;             // A/B frag: hi half-wave holds K+2,K+3

  v8f acc[4][4];
#pragma unroll
  for (int i = 0; i < 4; ++i)
#pragma unroll
    for (int j = 0; j < 4; ++j)
#pragma unroll
      for (int e = 0; e < 8; ++e) acc[i][j][e] = 0.0f;

  // Per-tile loading: 256 rows x 16 floats = 1024 float4 per operand;
  // 512 threads -> 2 async b128 per operand per thread.
  // f = tid + i*512 : row = f>>2 (4 float4 per row), col = (f&3)*4.
  const int r0 = (tid + 0 * THREADS) >> 2, c0 = ((tid + 0 * THREADS) & 3) * 4;
  const int r1 = (tid + 1 * THREADS) >> 2, c1 = ((tid + 1 * THREADS) & 3) * 4;

  const float* Xrow0 = X    + (size_t)(m0 + r0) * K_DIM + c0;
  const float* Xrow1 = X    + (size_t)(m0 + r1) * K_DIM + c1;
  const float* Wrow0 = Weff + (size_t)(n0 + r0) * K_DIM + c0;
  const float* Wrow1 = Weff + (size_t)(n0 + r1) * K_DIM + c1;

  // --- prologue: async-load K-tile 0 into buffer 0 ---
  async_load_b128(Xrow0, &xs[0][r0][c0]);
  async_load_b128(Xrow1, &xs[0][r1][c1]);
  async_load_b128(Wrow0, &wt[0][r0][c0]);
  async_load_b128(Wrow1, &wt[0][r1][c1]);
  wait_async0();
  __syncthreads();

  const int NT = K_DIM / BK;             // 128 K-tiles
  for (int kt = 0; kt < NT; ++kt) {
    const int cur = kt & 1, nxt = cur ^ 1;

    // issue async loads for the next K-tile straight into LDS buffer nxt
    if (kt + 1 < NT) {
      const int kn = (kt + 1) * BK;
      async_load_b128(Xrow0 + kn, &xs[nxt][r0][c0]);
      async_load_b128(Xrow1 + kn, &xs[nxt][r1][c1]);
      async_load_b128(Wrow0 + kn, &wt[nxt][r0][c0]);
      async_load_b128(Wrow1 + kn, &wt[nxt][r1][c1]);
    }

    // compute on LDS buffer cur: 4 k-steps of K=4, 16 WMMAs each
#pragma unroll
    for (int ks = 0; ks < BK; ks += 4) {
      v2f af[4], bf[4];
#pragma unroll
      for (int mf = 0; mf < 4; ++mf)
        af[mf] = *(const v2f*)&xs[cur][waveM + mf * 16 + l15][ks + khalf];
#pragma unroll
      for (int nf = 0; nf < 4; ++nf)
        bf[nf] = *(const v2f*)&wt[cur][waveN + nf * 16 + l15][ks + khalf];
#pragma unroll
      for (int mf = 0; mf < 4; ++mf)
#pragma unroll
        for (int nf = 0; nf < 4; ++nf)
          acc[mf][nf] = __builtin_amdgcn_wmma_f32_16x16x4_f32(
              false, af[mf], false, bf[nf], (short)0, acc[mf][nf], false, false);
    }

    // my async loads into nxt done + everyone finished reading cur
    wait_async0();
    __syncthreads();
  }

  // --- epilogue: + bias, store ---
  float bvals[4];
#pragma unroll
  for (int nf = 0; nf < 4; ++nf)
    bvals[nf] = bias[n0 + waveN + nf * 16 + l15];

#pragma unroll
  for (int mf = 0; mf < 4; ++mf) {
#pragma unroll
    for (int nf = 0; nf < 4; ++nf) {
      const int n = n0 + waveN + nf * 16 + l15;
#pragma unroll
      for (int i = 0; i < 8; ++i) {      // C VGPR i: rows i (lanes 0-15) / i+8 (16-31)
        const int m = m0 + waveM + mf * 16 + hi * 8 + i;
        out[(size_t)m * N_DIM + n] = acc[mf][nf][i] + bvals[nf];
      }
    }
  }
}

// ---------------------------------------------------------------------------
extern "C" void kernel_launch(void* const* d_in, const int* in_sizes, int n_in,
                              void* d_out, int out_size, void* d_ws, size_t ws_size,
                              hipStream_t stream) {
  const float* x    = (const float*)d_in[0];   // [4,4096,2048]
  const float* W    = (const float*)d_in[1];   // [2048,2048]
  const float* b    = (const float*)d_in[2];   // [2048]
  const float* A    = (const float*)d_in[3];   // [16,2048]
  const float* Bm   = (const float*)d_in[4];   // [2048,16]
  float*       out  = (float*)d_out;           // [4,4096,2048]
  float*       Weff = (float*)d_ws;            // needs 2048*2048*4 = 16 MiB

  (void)in_sizes; (void)n_in; (void)out_size; (void)ws_size;

  // 1) W_eff = W + B@A  (folds the rank-16 LoRA update into the weight)
  weff_kernel<<<(N_DIM * K_DIM / 4) / 256, 256, 0, stream>>>(W, A, Bm, Weff);

  // 2) out = x @ W_eff^T + b  (fp32 WMMA GEMM, 256x256 block tiles, async LDS)
  lora_gemm_kernel<<<dim3(N_DIM / BN, M_DIM / BM), THREADS, 0, stream>>>(x, Weff, b, out);
}